// Net_81793357185728
// MI455X (gfx1250) — compile-verified
//
#include <hip/hip_runtime.h>
#include <hip/hip_bf16.h>
#include <math.h>

typedef float v2f __attribute__((ext_vector_type(2)));
typedef float v8f __attribute__((ext_vector_type(8)));

#define LB __launch_bounds__(256)

// ---------------------------------------------------------------------------
// Zero fill (graph-safe; avoids hipMemsetAsync)
// ---------------------------------------------------------------------------
__global__ LB void zero_kernel(float* __restrict__ p, size_t n) {
  size_t i = (size_t)blockIdx.x * blockDim.x + threadIdx.x;
  size_t stride = (size_t)gridDim.x * blockDim.x;
  for (; i < n; i += stride) p[i] = 0.0f;
}

// ---------------------------------------------------------------------------
// Per-target w_mul stats sentinels.
//  wp_max: int-ordered atomicMax over w>=0, init -inf
//  wp_min: int-ordered atomicMin over w>=0, init +inf
//  wn_max: uint-ordered atomicMin over w<0  (least negative), init -inf bits
//  wn_min: uint-ordered atomicMax over w<0  (most negative),  init -0.0 bits
// ---------------------------------------------------------------------------
__global__ LB void init_stats_kernel(unsigned* __restrict__ wp_max,
                                     unsigned* __restrict__ wp_min,
                                     unsigned* __restrict__ wn_max,
                                     unsigned* __restrict__ wn_min, int N) {
  int i = blockIdx.x * blockDim.x + threadIdx.x;
  if (i < N) {
    wp_max[i] = 0xFF800000u;
    wp_min[i] = 0x7F800000u;
    wn_max[i] = 0xFF800000u;
    wn_min[i] = 0x80000000u;
  }
}

__global__ LB void edge_stats_kernel(const int* __restrict__ ei,
                                     const float* __restrict__ wmul,
                                     unsigned* __restrict__ wp_max,
                                     unsigned* __restrict__ wp_min,
                                     unsigned* __restrict__ wn_max,
                                     unsigned* __restrict__ wn_min, int E) {
  int e = blockIdx.x * blockDim.x + threadIdx.x;
  if (e >= E) return;
  int t = ei[E + e];           // tgt = edge_index[1][e]
  float w = wmul[e];
  if (w >= 0.0f) {
    atomicMax((int*)(wp_max + t), __float_as_int(w));
    atomicMin((int*)(wp_min + t), __float_as_int(w));
  } else {
    atomicMin(wn_max + t, __float_as_uint(w));
    atomicMax(wn_min + t, __float_as_uint(w));
  }
}

// ---------------------------------------------------------------------------
// u[k] = sum_j lrelu(w1[j]) * w2[k,j]   (edge weight vector for w_mul >= 0)
// v[k] = sum_j  g(w1[j])   * w2[k,j]   (w_mul < 0),  g(t) = -lrelu(-t)
// (w2 bias cancels in the segment softmax, so it is dropped.)
// ---------------------------------------------------------------------------
__global__ LB void uv_kernel(const float* __restrict__ w1,
                             const float* __restrict__ w2,
                             float* __restrict__ u, float* __restrict__ v, int Kd) {
  int k = blockIdx.x * blockDim.x + threadIdx.x;
  if (k >= Kd) return;
  float au = 0.0f, av = 0.0f;
  for (int j = 0; j < Kd; ++j) {
    float t = w1[j];
    float lr = (t > 0.0f) ? t : 0.2f * t;
    float g  = (t > 0.0f) ? 0.2f * t : t;
    float wk = w2[(size_t)k * Kd + j];
    au += lr * wk;
    av += g * wk;
  }
  u[k] = au;
  v[k] = av;
}

// ---------------------------------------------------------------------------
// Cout[M,Ncols] = A[M,K] @ W[Ncols,K]^T + bias, f32 WMMA 16x16x4.
// Block covers 32 rows; each wave owns one 16-col tile and accumulates TWO
// row tiles sharing a single B fragment (2 independent WMMA chains per
// load-wait, half the B traffic). K-loop is software-pipelined: fragments
// for k+4 are in flight while the k-step WMMAs execute.
// Out-of-range columns/rows: reads are clamped in-bounds (finite garbage can
// only affect its own masked row/column), stores are guarded.
// ---------------------------------------------------------------------------
__global__ LB void gemm_bias_wmma(const float* __restrict__ A,
                                  const float* __restrict__ W,
                                  const float* __restrict__ bias,
                                  float* __restrict__ Cout,
                                  int M, int K, int Ncols) {
  const int wave = threadIdx.x >> 5;
  const int lane = threadIdx.x & 31;
  const int half = lane >> 4;       // lanes 16..31 hold K+2..K+3 / rows M+8..15
  const int l    = lane & 15;
  const int row0 = blockIdx.x << 5; // 32 rows per block

  int ra0 = row0 + l;       if (ra0 >= M) ra0 = M - 1;  // clamped reads
  int ra1 = row0 + 16 + l;  if (ra1 >= M) ra1 = M - 1;
  const float* arow0 = A + (size_t)ra0 * K;
  const float* arow1 = A + (size_t)ra1 * K;

  const int nColTiles = (Ncols + 15) >> 4;
  for (int ct = wave; ct < nColTiles; ct += 8) {
    const int col = (ct << 4) + l;
    const bool cok = (col < Ncols);
    const float* wrow = W + (size_t)(cok ? col : 0) * K;  // clamped, finite

    v8f acc0 = {};
    v8f acc1 = {};
    const int koff = half * 2;

    // prologue
    v2f a0, a1, b;
    a0.x = arow0[koff];     a0.y = arow0[koff + 1];
    a1.x = arow1[koff];     a1.y = arow1[koff + 1];
    b.x  = wrow[koff];      b.y  = wrow[koff + 1];

    for (int k0 = 4; k0 < K; k0 += 4) {
      const int kb = k0 + koff;
      v2f na0, na1, nb;                      // next fragments in flight
      na0.x = arow0[kb];  na0.y = arow0[kb + 1];
      na1.x = arow1[kb];  na1.y = arow1[kb + 1];
      nb.x  = wrow[kb];   nb.y  = wrow[kb + 1];
      acc0 = __builtin_amdgcn_wmma_f32_16x16x4_f32(false, a0, false, b, (short)0,
                                                   acc0, false, false);
      acc1 = __builtin_amdgcn_wmma_f32_16x16x4_f32(false, a1, false, b, (short)0,
                                                   acc1, false, false);
      a0 = na0; a1 = na1; b = nb;
    }
    acc0 = __builtin_amdgcn_wmma_f32_16x16x4_f32(false, a0, false, b, (short)0,
                                                 acc0, false, false);
    acc1 = __builtin_amdgcn_wmma_f32_16x16x4_f32(false, a1, false, b, (short)0,
                                                 acc1, false, false);

    if (cok) {
      const float bv = bias[col];
#pragma unroll
      for (int r = 0; r < 8; ++r) {
        const int rowA = row0 + r + half * 8;
        const int rowB = rowA + 16;
        if (rowA < M) Cout[(size_t)rowA * Ncols + col] = acc0[r] + bv;
        if (rowB < M) Cout[(size_t)rowB * Ncols + col] = acc1[r] + bv;
      }
    }
  }
}

// ---------------------------------------------------------------------------
// Edge softmax passes. One wave per edge; lanes stride the feature dim so
// atomics per wave hit consecutive addresses of one L2-resident row.
// m[t,k] is reconstructed from the 4 per-target w_mul stats (no E*K max pass).
// ---------------------------------------------------------------------------
__device__ __forceinline__ float seg_m(float uk, float vk, float pmax, float pmin,
                                       float nmax, float nmin, bool posp, bool negp) {
  float mp = (uk >= 0.0f ? pmax : pmin) * uk;
  float mn = (vk >= 0.0f ? nmax : nmin) * vk;
  float m = -3.0e38f;
  if (posp) m = mp;
  if (negp) m = fmaxf(m, mn);
  return m;
}

__global__ LB void edge_sum_kernel(const int* __restrict__ ei,
                                   const float* __restrict__ wmul,
                                   const float* __restrict__ u,
                                   const float* __restrict__ v,
                                   const float* __restrict__ wp_max,
                                   const float* __restrict__ wp_min,
                                   const float* __restrict__ wn_max,
                                   const float* __restrict__ wn_min,
                                   float* __restrict__ s, int E, int Kd) {
  extern __shared__ float sm[];
  float* su = sm;
  float* sv = sm + Kd;
  for (int i = threadIdx.x; i < Kd; i += blockDim.x) { su[i] = u[i]; sv[i] = v[i]; }
  __syncthreads();

  const int wave = threadIdx.x >> 5;
  const int lane = threadIdx.x & 31;
  const int e = blockIdx.x * 8 + wave;
  if (e >= E) return;

  const int t = ei[E + e];
  const float w = wmul[e];
  const float pmax = wp_max[t], pmin = wp_min[t];
  const float nmax = wn_max[t], nmin = wn_min[t];
  const bool posp = (__float_as_uint(pmax) != 0xFF800000u);
  const bool negp = (__float_as_uint(nmax) != 0xFF800000u);
  const bool wpos = (w >= 0.0f);
  float* srow = s + (size_t)t * Kd;

  for (int k = lane; k < Kd; k += 32) {
    float uk = su[k], vk = sv[k];
    float m = seg_m(uk, vk, pmax, pmin, nmax, nmin, posp, negp);
    float c = wpos ? uk : vk;
    atomicAdd(srow + k, __expf(w * c - m));
  }
}

__global__ LB void edge_msg_kernel(const int* __restrict__ ei,
                                   const float* __restrict__ wmul,
                                   const float* __restrict__ u,
                                   const float* __restrict__ v,
                                   const float* __restrict__ wp_max,
                                   const float* __restrict__ wp_min,
                                   const float* __restrict__ wn_max,
                                   const float* __restrict__ wn_min,
                                   const float* __restrict__ s,
                                   const float* __restrict__ x,
                                   float* __restrict__ out, int E, int Kd) {
  extern __shared__ float sm[];
  float* su = sm;
  float* sv = sm + Kd;
  for (int i = threadIdx.x; i < Kd; i += blockDim.x) { su[i] = u[i]; sv[i] = v[i]; }
  __syncthreads();

  const int wave = threadIdx.x >> 5;
  const int lane = threadIdx.x & 31;
  const int e = blockIdx.x * 8 + wave;
  if (e >= E) return;

  const int src = ei[e];
  const int t = ei[E + e];
  const float w = wmul[e];
  const float pmax = wp_max[t], pmin = wp_min[t];
  const float nmax = wn_max[t], nmin = wn_min[t];
  const bool posp = (__float_as_uint(pmax) != 0xFF800000u);
  const bool negp = (__float_as_uint(nmax) != 0xFF800000u);
  const bool wpos = (w >= 0.0f);
  const float* srow = s + (size_t)t * Kd;
  const float* xrow = x + (size_t)src * Kd;
  float* orow = out + (size_t)t * Kd;

  for (int k = lane; k < Kd; k += 32) {
    float uk = su[k], vk = sv[k];
    float m = seg_m(uk, vk, pmax, pmin, nmax, nmin, posp, negp);
    float c = wpos ? uk : vk;
    float ev = __expf(w * c - m);
    float coef = ev / (srow[k] + 1e-16f);
    atomicAdd(orow + k, coef * xrow[k]);
  }
}

// ---------------------------------------------------------------------------
__global__ LB void selu_kernel(float* __restrict__ p, size_t n) {
  const float scale = 1.0507009873554805f;
  const float alpha = 1.6732632423543772f;
  size_t i = (size_t)blockIdx.x * blockDim.x + threadIdx.x;
  size_t stride = (size_t)gridDim.x * blockDim.x;
  for (; i < n; i += stride) {
    float x = p[i];
    p[i] = (x > 0.0f) ? scale * x : scale * alpha * (__expf(x) - 1.0f);
  }
}

__global__ LB void logsoftmax_kernel(const float* __restrict__ logits,
                                     float* __restrict__ outp, int N, int C) {
  int i = blockIdx.x * blockDim.x + threadIdx.x;
  if (i >= N) return;
  const float* r = logits + (size_t)i * C;
  float m = -3.0e38f;
  for (int j = 0; j < C; ++j) m = fmaxf(m, r[j]);
  float ssum = 0.0f;
  for (int j = 0; j < C; ++j) ssum += __expf(r[j] - m);
  float ls = logf(ssum);
  float* o = outp + (size_t)i * C;
  for (int j = 0; j < C; ++j) o[j] = r[j] - m - ls;
}

// ---------------------------------------------------------------------------
extern "C" void kernel_launch(void* const* d_in, const int* in_sizes, int n_in,
                              void* d_out, int out_size, void* d_ws, size_t ws_size,
                              hipStream_t stream) {
  (void)n_in; (void)out_size; (void)ws_size;
  const float* x    = (const float*)d_in[0];
  const int*   ei   = (const int*)d_in[1];
  const float* wmul = (const float*)d_in[2];
  const float* l1w  = (const float*)d_in[3];
  const float* l1b  = (const float*)d_in[4];
  const float* w1a  = (const float*)d_in[5];
  const float* w2a  = (const float*)d_in[6];
  // d_in[7] = conv1_w2_b: cancels in segment softmax -> unused
  const float* l2w  = (const float*)d_in[8];
  const float* l2b  = (const float*)d_in[9];
  const float* w1b  = (const float*)d_in[10];
  const float* w2b  = (const float*)d_in[11];
  // d_in[12] = conv2_w2_b: unused (same reason)

  const int H = in_sizes[4];             // 128
  const int C = in_sizes[9];             // 40
  const int F = in_sizes[3] / H;         // 128
  const int N = in_sizes[0] / F;         // 50000
  const int E = in_sizes[2];             // 800000
  const size_t NH = (size_t)N * H;
  const size_t NC = (size_t)N * C;

  float* ws = (float*)d_ws;
  float* x1 = ws;                 // [N,H]
  float* s1 = x1 + NH;            // [N,H]
  float* o1 = s1 + NH;            // [N,H] layer-1 output, selu'd in place
  float* x2 = o1 + NH;            // [N,C]
  float* s2 = x2 + NC;            // [N,C]
  float* wp_max = s2 + NC;        // [N] stats (shared by both layers)
  float* wp_min = wp_max + N;
  float* wn_max = wp_min + N;
  float* wn_min = wn_max + N;
  float* u1 = wn_min + N;         // [H]
  float* v1 = u1 + H;             // [H]
  float* u2 = v1 + H;             // [C]
  float* v2 = u2 + C;             // [C]

  float* outLS  = (float*)d_out;  // [N,C] log_softmax
  float* logits = outLS + NC;     // [N,C] logits (accumulated directly)

  const int T = 256;
  // zero accumulators + output
  zero_kernel<<<2048, T, 0, stream>>>(s1, NH);
  zero_kernel<<<2048, T, 0, stream>>>(o1, NH);
  zero_kernel<<<2048, T, 0, stream>>>(s2, NC);
  zero_kernel<<<2048, T, 0, stream>>>((float*)d_out, 2 * NC);

  // per-target w_mul stats (layer-independent)
  init_stats_kernel<<<(N + T - 1) / T, T, 0, stream>>>(
      (unsigned*)wp_max, (unsigned*)wp_min, (unsigned*)wn_max, (unsigned*)wn_min, N);
  edge_stats_kernel<<<(E + T - 1) / T, T, 0, stream>>>(
      ei, wmul, (unsigned*)wp_max, (unsigned*)wp_min, (unsigned*)wn_max,
      (unsigned*)wn_min, E);

  // ----- layer 1 -----
  uv_kernel<<<(H + T - 1) / T, T, 0, stream>>>(w1a, w2a, u1, v1, H);
  gemm_bias_wmma<<<(N + 31) / 32, T, 0, stream>>>(x, l1w, l1b, x1, N, F, H);
  edge_sum_kernel<<<(E + 7) / 8, T, 2 * H * sizeof(float), stream>>>(
      ei, wmul, u1, v1, wp_max, wp_min, wn_max, wn_min, s1, E, H);
  edge_msg_kernel<<<(E + 7) / 8, T, 2 * H * sizeof(float), stream>>>(
      ei, wmul, u1, v1, wp_max, wp_min, wn_max, wn_min, s1, x1, o1, E, H);
  selu_kernel<<<2048, T, 0, stream>>>(o1, NH);

  // ----- layer 2 -----
  uv_kernel<<<(C + T - 1) / T, T, 0, stream>>>(w1b, w2b, u2, v2, C);
  gemm_bias_wmma<<<(N + 31) / 32, T, 0, stream>>>(o1, l2w, l2b, x2, N, H, C);
  edge_sum_kernel<<<(E + 7) / 8, T, 2 * C * sizeof(float), stream>>>(
      ei, wmul, u2, v2, wp_max, wp_min, wn_max, wn_min, s2, E, C);
  edge_msg_kernel<<<(E + 7) / 8, T, 2 * C * sizeof(float), stream>>>(
      ei, wmul, u2, v2, wp_max, wp_min, wn_max, wn_min, s2, x2, logits, E, C);

  logsoftmax_kernel<<<(N + T - 1) / T, T, 0, stream>>>(logits, outLS, N, C);
}